// LiteMSA_29042568856119
// MI455X (gfx1250) — compile-verified
//
#include <hip/hip_runtime.h>

// LiteMSA for MI455X (gfx1250, wave32). fp32 end-to-end; big GEMMs use
// V_WMMA_F32_16X16X4_F32 (exact fp32) fed from LDS tiles staged by the
// Tensor Data Mover (tensor_load_to_lds, TENSORcnt), double-buffered.
// Problem is HBM-bandwidth bound (~1.5 GB traffic vs 63 GFLOP).

typedef __attribute__((ext_vector_type(2))) float v2f;
typedef __attribute__((ext_vector_type(8))) float v8f;
typedef __attribute__((ext_vector_type(4))) unsigned int v4u;
typedef __attribute__((ext_vector_type(8))) int v8i;
typedef __attribute__((ext_vector_type(4))) int v4i;

#define HW   4096   // 64*64
#define BANK 16     // batches
#define KB   32     // K-block staged per TDM transfer
#define NT   128    // N columns per block (8 waves x 16)

// ---------------------------------------------------------------------------
// Issue one 2-D TDM tile load: tile_rows x tile_cols fp32 elements from a
// row-major matrix (row stride = row_stride_elems) into contiguous LDS at
// byte offset lds_off.  D# packing per CDNA5 ISA §8.3/8.4 (2-D: groups 2/3
// zero, tile_dim2=0).  EXEC-independent; issued once per executing wave.
// This toolchain exposes the 6-arg builtin: (g0, g1, g2, g3, g4ext, cpol).
// ---------------------------------------------------------------------------
__device__ __forceinline__ void tdm_load_2d(unsigned lds_off, const void* gaddr,
                                            unsigned tile_cols, unsigned tile_rows,
                                            unsigned row_stride_elems)
{
    const unsigned long long ga = (unsigned long long)(uintptr_t)gaddr;
    const unsigned td0 = row_stride_elems;   // tensor dim0: full row length
    const unsigned td1 = 1u << 20;           // tall tensor: OOB never triggers
    v4u g0 = { 1u,                                           // count=1, user D#
               lds_off,                                      // lds_addr (bytes)
               (unsigned)ga,                                 // global_addr[31:0]
               (unsigned)((ga >> 32) & 0x01ffffffu) | (2u << 30) }; // [56:32]|type=2
    v8i g1 = { (int)0x00020000u,                             // data_size=4B, mask=0
               (int)((td0 & 0xffffu) << 16),                 // tensor_dim0 lo16
               (int)(((td0 >> 16) & 0xffffu) | ((td1 & 0xffffu) << 16)),
               (int)(((td1 >> 16) & 0xffffu) | ((tile_cols & 0xffffu) << 16)),
               (int)(tile_rows & 0xffffu),                   // tile_dim1; tile_dim2=0
               (int)row_stride_elems,                        // dim0_stride lo32
               0, 0 };
    v4i gz4 = { 0, 0, 0, 0 };
    v8i gz8 = { 0, 0, 0, 0, 0, 0, 0, 0 };
    __builtin_amdgcn_tensor_load_to_lds(g0, g1, gz4, gz4, gz8, 0);
}

// ---------------------------------------------------------------------------
// TDM-staged fp32 WMMA GEMM:  Y[b][m][n] = sum_k A[m][k] * X[b][k][n]
// block = 256 threads = 8 waves; block tile = 16 (M) x 128 (N).
// K consumed in KB=32 slabs, double-buffered in LDS via tensor_load_to_lds.
// grid.x = N/128, grid.y = M/16, grid.z = batch.
// Optional fused batchnorm epilogue (gamma != nullptr).
// ---------------------------------------------------------------------------
__global__ __launch_bounds__(256)
void wmma_gemm_bn(const float* __restrict__ A, const float* __restrict__ X,
                  float* __restrict__ Y, int Mdim, int Kdim,
                  const float* __restrict__ gamma, const float* __restrict__ beta,
                  const float* __restrict__ mean,  const float* __restrict__ var)
{
    __shared__ float XT[2][KB][NT];   // 32 KB: X slab (K x N)
    __shared__ float AT[2][16][KB];   //  4 KB: A slab (M x K)

    const int lane = threadIdx.x & 31;
    const int wave = threadIdx.x >> 5;
    const int b    = blockIdx.z;
    const int m0   = blockIdx.y * 16;
    const int nblk = blockIdx.x * NT;
    const int half = lane >> 4;      // K-pair selector (ISA 16x4 f32 A/B layout)
    const int l16  = lane & 15;      // row (A) / column (B,C,D)
    const int wcol = wave * 16 + l16;

    const float* Xb = X + (size_t)b * Kdim * HW + nblk;   // slab base, this block
    const float* Ab = A + (size_t)m0 * Kdim;              // A tile rows m0..m0+15
    const int NB = Kdim / KB;

    // Prime the pipeline: slab 0 in flight.
    if (wave == 0) {
        tdm_load_2d((unsigned)(uintptr_t)&XT[0][0][0], Xb,            NT, KB, HW);
        tdm_load_2d((unsigned)(uintptr_t)&AT[0][0][0], Ab,            KB, 16, Kdim);
    }

    v8f acc = {};
    for (int j = 0; j < NB; ++j) {
        const int cur = j & 1;
        if (wave == 0) {
            if (j + 1 < NB) {
                const int nxt = cur ^ 1;
                tdm_load_2d((unsigned)(uintptr_t)&XT[nxt][0][0],
                            Xb + (size_t)(j + 1) * KB * HW, NT, KB, HW);
                tdm_load_2d((unsigned)(uintptr_t)&AT[nxt][0][0],
                            Ab + (size_t)(j + 1) * KB,      KB, 16, Kdim);
                __builtin_amdgcn_s_wait_tensorcnt(2);  // slab j done (in-order)
            } else {
                __builtin_amdgcn_s_wait_tensorcnt(0);  // last slab done
            }
        }
        __syncthreads();                               // slab j visible to all waves

        #pragma unroll
        for (int kk = 0; kk < KB; kk += 4) {
            const int ka = kk + half * 2;
            v2f av, bv;
            // A 16x4: lanes 0-15 hold K={k,k+1}, lanes 16-31 hold K={k+2,k+3}
            av.x = AT[cur][l16][ka];
            av.y = AT[cur][l16][ka + 1];
            // B 4x16: same K pairing, N striped across 16 lanes
            bv.x = XT[cur][ka][wcol];
            bv.y = XT[cur][ka + 1][wcol];
            acc = __builtin_amdgcn_wmma_f32_16x16x4_f32(
                false, av, false, bv, (short)0, acc, false, false);
        }
        __syncthreads();                               // done reading before overwrite
    }

    // C/D 16x16 f32: VGPR r -> M = m0+r (lanes 0-15) / m0+r+8 (lanes 16-31)
    float* Yb = Y + (size_t)b * Mdim * HW;
    #pragma unroll
    for (int r = 0; r < 8; ++r) {
        const int row = m0 + r + half * 8;
        float v = acc[r];
        if (gamma) {
            const float inv = gamma[row] * rsqrtf(var[row] + 1e-5f);
            v = v * inv + (beta[row] - mean[row] * inv);
        }
        Yb[(size_t)row * HW + (nblk + wcol)] = v;
    }
}

// ---------------------------------------------------------------------------
// Fused depthwise 3x3 (SAME, zero pad, cross-correlation) + grouped 1x1
// (96 groups of 8). One pixel per thread; grid (HW/256, 96, B).
// ---------------------------------------------------------------------------
__global__ __launch_bounds__(256)
void dwpw_kernel(const float* __restrict__ qkv, const float* __restrict__ wdw,
                 const float* __restrict__ wpw, float* __restrict__ agg)
{
    const int n = blockIdx.x * 256 + threadIdx.x;
    const int g = blockIdx.y;
    const int b = blockIdx.z;
    const int x = n & 63, y = n >> 6;

    float dwv[8];
    #pragma unroll
    for (int i = 0; i < 8; ++i) {
        const int ch = g * 8 + i;
        const float* p = qkv + ((size_t)b * 768 + ch) * HW;
        const float* w = wdw + ch * 9;
        float s = 0.f;
        #pragma unroll
        for (int ky = 0; ky < 3; ++ky) {
            const int yy = y + ky - 1;
            if ((unsigned)yy < 64u) {
                #pragma unroll
                for (int kx = 0; kx < 3; ++kx) {
                    const int xx = x + kx - 1;
                    if ((unsigned)xx < 64u)
                        s = fmaf(w[ky * 3 + kx], p[yy * 64 + xx], s);
                }
            }
        }
        dwv[i] = s;
    }
    #pragma unroll
    for (int o = 0; o < 8; ++o) {
        const float* w = wpw + (size_t)(g * 8 + o) * 8;
        float s = 0.f;
        #pragma unroll
        for (int i = 0; i < 8; ++i) s = fmaf(w[i], dwv[i], s);
        agg[((size_t)b * 768 + g * 8 + o) * HW + n] = s;
    }
}

// ---------------------------------------------------------------------------
// kv[b,h,d,e] = sum_n relu(k[n,d]) * {v[n,e] | 1}.  grid (64 heads, B),
// 256 threads; register accumulators -> wave32 shuffle reduce -> LDS atomics.
// ---------------------------------------------------------------------------
__global__ __launch_bounds__(256)
void kv_kernel(const float* __restrict__ qkv, const float* __restrict__ agg,
               float* __restrict__ kvb)
{
    const int h  = blockIdx.x;
    const int b  = blockIdx.y;
    const int cb = h * 24;
    const float* src = (h < 32)
        ? qkv + ((size_t)b * 768 + cb) * HW
        : agg + ((size_t)b * 768 + (cb - 768)) * HW;

    float acc[72];
    #pragma unroll
    for (int i = 0; i < 72; ++i) acc[i] = 0.f;

    for (int n = threadIdx.x; n < HW; n += 256) {
        float kk[8], vv[8];
        #pragma unroll
        for (int d = 0; d < 8; ++d) kk[d] = fmaxf(src[(size_t)(8 + d) * HW + n], 0.f);
        #pragma unroll
        for (int e = 0; e < 8; ++e) vv[e] = src[(size_t)(16 + e) * HW + n];
        #pragma unroll
        for (int d = 0; d < 8; ++d) {
            #pragma unroll
            for (int e = 0; e < 8; ++e)
                acc[d * 9 + e] = fmaf(kk[d], vv[e], acc[d * 9 + e]);
            acc[d * 9 + 8] += kk[d];   // v augmented with ones column
        }
    }

    __shared__ float s[72];
    if (threadIdx.x < 72) s[threadIdx.x] = 0.f;
    __syncthreads();

    const int lane = threadIdx.x & 31;          // wave32
    #pragma unroll
    for (int i = 0; i < 72; ++i) {
        float v = acc[i];
        #pragma unroll
        for (int off = 16; off > 0; off >>= 1) v += __shfl_down(v, off, 32);
        if (lane == 0) atomicAdd(&s[i], v);
    }
    __syncthreads();
    if (threadIdx.x < 72)
        kvb[((size_t)b * 64 + h) * 72 + threadIdx.x] = s[threadIdx.x];
}

// ---------------------------------------------------------------------------
// att[b, h*8+d, n] = (sum_d' relu(q[n,d']) kv[d',d]) / (sum_d' q kv[d',8] + eps)
// grid (HW/256, 64, B); kv tile staged in LDS.
// ---------------------------------------------------------------------------
__global__ __launch_bounds__(256)
void att_kernel(const float* __restrict__ qkv, const float* __restrict__ agg,
                const float* __restrict__ kvb, float* __restrict__ att)
{
    const int h = blockIdx.y;
    const int b = blockIdx.z;
    const int n = blockIdx.x * 256 + threadIdx.x;

    __shared__ float skv[72];
    if (threadIdx.x < 72)
        skv[threadIdx.x] = kvb[((size_t)b * 64 + h) * 72 + threadIdx.x];
    __syncthreads();

    const int cb = h * 24;
    const float* src = (h < 32)
        ? qkv + ((size_t)b * 768 + cb) * HW
        : agg + ((size_t)b * 768 + (cb - 768)) * HW;

    float q[8];
    #pragma unroll
    for (int d = 0; d < 8; ++d) q[d] = fmaxf(src[(size_t)d * HW + n], 0.f);

    float out[9];
    #pragma unroll
    for (int e = 0; e < 9; ++e) {
        float s = 0.f;
        #pragma unroll
        for (int d = 0; d < 8; ++d) s = fmaf(q[d], skv[d * 9 + e], s);
        out[e] = s;
    }
    const float den = out[8] + 1e-15f;
    #pragma unroll
    for (int d = 0; d < 8; ++d)
        att[((size_t)b * 512 + h * 8 + d) * HW + n] = out[d] / den;
}

// ---------------------------------------------------------------------------
extern "C" void kernel_launch(void* const* d_in, const int* in_sizes, int n_in,
                              void* d_out, int out_size, void* d_ws, size_t ws_size,
                              hipStream_t stream)
{
    const float* x      = (const float*)d_in[0];
    const float* w_qkv  = (const float*)d_in[1];
    const float* w_dw   = (const float*)d_in[2];
    const float* w_pw   = (const float*)d_in[3];
    const float* w_proj = (const float*)d_in[4];
    const float* gamma  = (const float*)d_in[5];
    const float* beta   = (const float*)d_in[6];
    const float* mean   = (const float*)d_in[7];
    const float* var    = (const float*)d_in[8];
    float* out = (float*)d_out;
    float* ws  = (float*)d_ws;

    // workspace layout (floats)
    float* qkv = ws;                                   // 16*768*4096
    float* agg = qkv + (size_t)BANK * 768 * HW;        // 16*768*4096
    float* kvb = agg + (size_t)BANK * 768 * HW;        // 16*64*72
    float* att = kvb + (size_t)BANK * 64 * 72;         // 16*512*4096

    const dim3 blk(256);

    // 1) qkv = w_qkv (768x256) @ x (256x4096) per batch      [TDM + WMMA f32]
    wmma_gemm_bn<<<dim3(HW / NT, 48, BANK), blk, 0, stream>>>(
        w_qkv, x, qkv, 768, 256, nullptr, nullptr, nullptr, nullptr);

    // 2) agg = grouped-1x1( depthwise-3x3( qkv ) )
    dwpw_kernel<<<dim3(HW / 256, 96, BANK), blk, 0, stream>>>(qkv, w_dw, w_pw, agg);

    // 3) kv reduction per (b, head)
    kv_kernel<<<dim3(64, BANK), blk, 0, stream>>>(qkv, agg, kvb);

    // 4) attention output (512 channels)
    att_kernel<<<dim3(HW / 256, 64, BANK), blk, 0, stream>>>(qkv, agg, kvb, att);

    // 5) y = w_proj (256x512) @ att (512x4096) per batch + BN [TDM + WMMA f32]
    wmma_gemm_bn<<<dim3(HW / NT, 16, BANK), blk, 0, stream>>>(
        w_proj, att, out, 256, 512, gamma, beta, mean, var);
}